// CamBase_44375602102919
// MI455X (gfx1250) — compile-verified
//
#include <hip/hip_runtime.h>
#include <hip/hip_bf16.h>

typedef float v2f __attribute__((ext_vector_type(2)));
typedef float v8f __attribute__((ext_vector_type(8)));

#define B_  4
#define C_  80
#define D_  118
#define O_  198     // D_ + C_
#define K_  80
#define FH_ 32
#define FW_ 88
#define P_  2816    // FH_*FW_
#define NX_ 180
#define NY_ 80

// ---------------------------------------------------------------------------
// Kernel 0: 4x4 matrix inverse (one thread per matrix). Layout-agnostic
// adjugate method: feeding row-major data yields row-major inverse.
// ---------------------------------------------------------------------------
__global__ void CamBase_inv4x4(const float* __restrict__ L,
                               float* __restrict__ out, int n) {
    int t = blockIdx.x * blockDim.x + threadIdx.x;
    if (t >= n) return;
    const float* m = L + t * 16;
    float inv[16];
    inv[0]  =  m[5]*m[10]*m[15] - m[5]*m[11]*m[14] - m[9]*m[6]*m[15] + m[9]*m[7]*m[14] + m[13]*m[6]*m[11] - m[13]*m[7]*m[10];
    inv[4]  = -m[4]*m[10]*m[15] + m[4]*m[11]*m[14] + m[8]*m[6]*m[15] - m[8]*m[7]*m[14] - m[12]*m[6]*m[11] + m[12]*m[7]*m[10];
    inv[8]  =  m[4]*m[9]*m[15]  - m[4]*m[11]*m[13] - m[8]*m[5]*m[15] + m[8]*m[7]*m[13] + m[12]*m[5]*m[11] - m[12]*m[7]*m[9];
    inv[12] = -m[4]*m[9]*m[14]  + m[4]*m[10]*m[13] + m[8]*m[5]*m[14] - m[8]*m[6]*m[13] - m[12]*m[5]*m[10] + m[12]*m[6]*m[9];
    inv[1]  = -m[1]*m[10]*m[15] + m[1]*m[11]*m[14] + m[9]*m[2]*m[15] - m[9]*m[3]*m[14] - m[13]*m[2]*m[11] + m[13]*m[3]*m[10];
    inv[5]  =  m[0]*m[10]*m[15] - m[0]*m[11]*m[14] - m[8]*m[2]*m[15] + m[8]*m[3]*m[14] + m[12]*m[2]*m[11] - m[12]*m[3]*m[10];
    inv[9]  = -m[0]*m[9]*m[15]  + m[0]*m[11]*m[13] + m[8]*m[1]*m[15] - m[8]*m[3]*m[13] - m[12]*m[1]*m[11] + m[12]*m[3]*m[9];
    inv[13] =  m[0]*m[9]*m[14]  - m[0]*m[10]*m[13] - m[8]*m[1]*m[14] + m[8]*m[2]*m[13] + m[12]*m[1]*m[10] - m[12]*m[2]*m[9];
    inv[2]  =  m[1]*m[6]*m[15]  - m[1]*m[7]*m[14]  - m[5]*m[2]*m[15] + m[5]*m[3]*m[14] + m[13]*m[2]*m[7]  - m[13]*m[3]*m[6];
    inv[6]  = -m[0]*m[6]*m[15]  + m[0]*m[7]*m[14]  + m[4]*m[2]*m[15] - m[4]*m[3]*m[14] - m[12]*m[2]*m[7]  + m[12]*m[3]*m[6];
    inv[10] =  m[0]*m[5]*m[15]  - m[0]*m[7]*m[13]  - m[4]*m[1]*m[15] + m[4]*m[3]*m[13] + m[12]*m[1]*m[7]  - m[12]*m[3]*m[5];
    inv[14] = -m[0]*m[5]*m[14]  + m[0]*m[6]*m[13]  + m[4]*m[1]*m[14] - m[4]*m[2]*m[13] - m[12]*m[1]*m[6]  + m[12]*m[2]*m[5];
    inv[3]  = -m[1]*m[6]*m[11]  + m[1]*m[7]*m[10]  + m[5]*m[2]*m[11] - m[5]*m[3]*m[10] - m[9]*m[2]*m[7]   + m[9]*m[3]*m[6];
    inv[7]  =  m[0]*m[6]*m[11]  - m[0]*m[7]*m[10]  - m[4]*m[2]*m[11] + m[4]*m[3]*m[10] + m[8]*m[2]*m[7]   - m[8]*m[3]*m[6];
    inv[11] = -m[0]*m[5]*m[11]  + m[0]*m[7]*m[9]   + m[4]*m[1]*m[11] - m[4]*m[3]*m[9]  - m[8]*m[1]*m[7]   + m[8]*m[3]*m[5];
    inv[15] =  m[0]*m[5]*m[10]  - m[0]*m[6]*m[9]   - m[4]*m[1]*m[10] + m[4]*m[2]*m[9]  + m[8]*m[1]*m[6]   - m[8]*m[2]*m[5];
    float det = m[0]*inv[0] + m[1]*inv[4] + m[2]*inv[8] + m[3]*inv[12];
    float id = 1.0f / det;
    float* o = out + t * 16;
    #pragma unroll
    for (int i = 0; i < 16; ++i) o[i] = inv[i] * id;
}

// ---------------------------------------------------------------------------
// Kernel 1: GEMM  out[b][p][o] = sum_k w[o][k] * feat[b][k][p] + bias[o]
// via v_wmma_f32_16x16x4_f32. One wave per 16(o) x 16(p) tile, 20 K-steps,
// fully unrolled, branch-free loads (A row index clamped: rows o>=198 of D
// are never stored, so their A data is don't-care; clamping avoids OOB).
// A (16x4 f32, 2 VGPRs): lane m = lane&15, k = 2*(lane>=16) + vgpr
// B (4x16 f32, 2 VGPRs): lane n = lane&15, k = 2*(lane>=16) + vgpr
// C/D (16x16 f32, 8 VGPRs): m = 8*(lane>=16) + v, n = lane&15
// ---------------------------------------------------------------------------
__global__ __launch_bounds__(32)
void CamBase_gemm_wmma(const float* __restrict__ w,
                       const float* __restrict__ bias,
                       const float* __restrict__ feat,
                       float* __restrict__ out) {
    const int pt   = blockIdx.x;       // 0..175
    const int ot   = blockIdx.y;       // 0..12
    const int b    = blockIdx.z;       // 0..3
    const int lane = threadIdx.x;      // 0..31
    const int half = lane >> 4;
    const int idx  = lane & 15;
    const int k0   = half * 2;
    const int obase = ot * 16;
    const int pbase = pt * 16;
    // clamp: don't-care data for tail rows, never stored
    const int mrow = min(obase + idx, O_ - 1);

    const float* __restrict__ wrow = w + mrow * K_ + k0;
    const float* __restrict__ fcol =
        feat + (size_t)b * K_ * P_ + (size_t)k0 * P_ + pbase + idx;

    v8f acc = {};
    #pragma unroll
    for (int kk = 0; kk < K_; kk += 4) {
        v2f a, bm;
        a.x  = wrow[kk];
        a.y  = wrow[kk + 1];
        bm.x = fcol[(size_t)kk * P_];
        bm.y = fcol[(size_t)(kk + 1) * P_];
        acc = __builtin_amdgcn_wmma_f32_16x16x4_f32(
                  /*neg_a=*/false, a, /*neg_b=*/false, bm,
                  /*c_mod=*/(short)0, acc, /*reuse_a=*/false, /*reuse_b=*/false);
    }

    // store transposed to [b][p][o] so the splat kernel reads contiguously
    const int p = pbase + idx;
    float* __restrict__ orow = out + ((size_t)b * P_ + p) * O_;
    #pragma unroll
    for (int v = 0; v < 8; ++v) {
        int o = obase + half * 8 + v;
        if (o < O_) orow[o] = acc[v] + bias[o];
    }
}

// ---------------------------------------------------------------------------
// Kernel 2: per-pixel softmax over 118 depth bins + frustum geometry +
// run-compressed voxel scatter. One 128-thread block per (b, pixel).
// Output layout: out[b][c][gy][gx]  (4, 80, 80, 180)
// ---------------------------------------------------------------------------
__global__ __launch_bounds__(128)
void CamBase_splat(const float* __restrict__ gemm_out,
                   const float* __restrict__ invmat,
                   float* __restrict__ out) {
    const int p = blockIdx.x;      // 0..2815
    const int b = blockIdx.y;      // 0..3
    const int t = threadIdx.x;     // 0..127

    __shared__ float sDepth[D_];
    __shared__ float sCtx[C_];
    __shared__ int   sKey[D_];
    __shared__ float sRed[128];
    __shared__ int   sRunKey[D_];
    __shared__ float sRunW[D_];
    __shared__ int   sCnt;
    if (t == 0) sCnt = 0;

    const float* __restrict__ row = gemm_out + ((size_t)b * P_ + p) * O_;
    float logit = -3.402823466e38f;
    if (t < D_)      logit = row[t];
    else if (t < O_) sCtx[t - D_] = row[t];

    // max-reduce
    sRed[t] = logit;
    __syncthreads();
    #pragma unroll
    for (int s = 64; s > 0; s >>= 1) {
        if (t < s) sRed[t] = fmaxf(sRed[t], sRed[t + s]);
        __syncthreads();
    }
    const float mx = sRed[0];
    __syncthreads();

    // exp + sum-reduce
    float e = (t < D_) ? __expf(logit - mx) : 0.0f;
    sRed[t] = e;
    __syncthreads();
    #pragma unroll
    for (int s = 64; s > 0; s >>= 1) {
        if (t < s) sRed[t] += sRed[t + s];
        __syncthreads();
    }
    const float inv_sum = 1.0f / sRed[0];
    __syncthreads();

    // geometry: voxel key per depth bin (-1 if culled)
    if (t < D_) {
        sDepth[t] = e * inv_sum;
        const float* __restrict__ M = invmat + b * 16;
        const int   h  = p / FW_;
        const int   ww = p - h * FW_;
        const float xs = (float)ww * (703.0f / 87.0f);   // linspace(0, 703, 88)
        const float ys = (float)h  * (255.0f / 31.0f);   // linspace(0, 255, 32)
        const float dv = 1.0f + 0.5f * (float)t;         // depth bin value
        // geom = inv @ [xs*dv, ys*dv, dv, 1]
        const float g0 = (M[0]*xs + M[1]*ys + M[2])  * dv + M[3];
        const float g1 = (M[4]*xs + M[5]*ys + M[6])  * dv + M[7];
        const float g2 = (M[8]*xs + M[9]*ys + M[10]) * dv + M[11];
        // truncation (astype(int32) semantics)
        const int gx = (int)( g0          / 0.4f);
        const int gy = (int)((g1 + 16.0f) / 0.4f);
        const int gz = (int)((g2 +  2.0f) / 9.6f);
        const bool kept = (gx >= 0) & (gx < NX_) & (gy >= 0) & (gy < NY_) & (gz == 0);
        sKey[t] = kept ? (gy * NX_ + gx) : -1;
    }
    __syncthreads();

    // compress consecutive depth bins mapping to the same voxel
    if (t < D_) {
        const int key = sKey[t];
        if (key >= 0 && (t == 0 || sKey[t - 1] != key)) {
            float wsum = sDepth[t];
            int j = t + 1;
            while (j < D_ && sKey[j] == key) { wsum += sDepth[j]; ++j; }
            const int pos = atomicAdd(&sCnt, 1);
            sRunKey[pos] = key;
            sRunW[pos]   = wsum;
        }
    }
    __syncthreads();

    // scatter: nRuns * 80 atomic f32 adds into L2-resident BEV grid
    const int nRuns = sCnt;
    float* __restrict__ ob = out + (size_t)b * C_ * NY_ * NX_;
    for (int i = t; i < nRuns * C_; i += 128) {
        const int r = i / C_;
        const int c = i - r * C_;
        atomicAdd(ob + (size_t)c * (NY_ * NX_) + sRunKey[r], sRunW[r] * sCtx[c]);
    }
}

// ---------------------------------------------------------------------------
extern "C" void kernel_launch(void* const* d_in, const int* in_sizes, int n_in,
                              void* d_out, int out_size, void* d_ws, size_t ws_size,
                              hipStream_t stream) {
    const float* feat      = (const float*)d_in[0];   // (4,1,80,32,88)
    const float* lidar2img = (const float*)d_in[1];   // (4,1,4,4)
    const float* w_dn      = (const float*)d_in[2];   // (198,80)
    const float* b_dn      = (const float*)d_in[3];   // (198,)
    float*       out       = (float*)d_out;           // (4,80,80,180)

    float* ws_inv = (float*)d_ws;                     // 64 floats
    float* ws_gem = ws_inv + 64;                      // 4*2816*198 floats (~8.9 MB)

    // invert camera matrices
    CamBase_inv4x4<<<1, 32, 0, stream>>>(lidar2img, ws_inv, B_);

    // WMMA GEMM + bias -> [b][p][o]
    CamBase_gemm_wmma<<<dim3(P_ / 16, (O_ + 15) / 16, B_), 32, 0, stream>>>(
        w_dn, b_dn, feat, ws_gem);

    // zero the BEV accumulator, then softmax + splat
    hipMemsetAsync(d_out, 0, (size_t)out_size * sizeof(float), stream);
    CamBase_splat<<<dim3(P_, B_), 128, 0, stream>>>(ws_gem, ws_inv, out);
}